// Gate_13941463843214
// MI455X (gfx1250) — compile-verified
//
#include <hip/hip_runtime.h>
#include <math.h>
#include <stdint.h>

// ---------------------------------------------------------------------------
// MoE gating kernel for MI455X (gfx1250, wave32).
//   logits[N,4] = x[N,64] @ W[4,64]^T   via V_WMMA_F32_16X16X4_F32 (chained K)
//   x-tile staged global->LDS with GLOBAL_LOAD_ASYNC_TO_LDS_B128 (ASYNCcnt)
//   top-2 + softmax + dense scatter     via VALU epilogue
// HBM-bound (~560 MB total): coalesced 128B async copies, WMMA does the GEMM,
// branch-free addressing keeps VGPRs low for occupancy / latency hiding.
// ---------------------------------------------------------------------------

typedef float v2f __attribute__((ext_vector_type(2)));
typedef float v8f __attribute__((ext_vector_type(8)));

#define EMBED 64
#define NEXP 4
#define ROWPAD 68          // 64 + 4 pad floats: conflict-free LDS A-reads
#define TOK_PER_BLOCK 128
#define TOK_PER_WAVE 16

__global__ __launch_bounds__(256, 6)
void moe_gate_wmma_kernel(const float* __restrict__ x,
                          const float* __restrict__ W,
                          float* __restrict__ weights_out,  // [N,4]
                          float* __restrict__ idx_out,      // [N,2] as float
                          int n_tokens) {
  __shared__ __align__(16) float xs[TOK_PER_BLOCK * ROWPAD];  // 34,816 B
  __shared__ float logits_s[TOK_PER_BLOCK * NEXP];            // 2 KB

  const int tid  = threadIdx.x;
  const int lane = tid & 31;
  const int wave = tid >> 5;
  const int blockTok = blockIdx.x * TOK_PER_BLOCK;

  // ---- stage x tile into LDS via async global->LDS copies (no dest VGPRs) ----
  {
    const uint64_t xbase = (uint64_t)(uintptr_t)x;
    const uint32_t lastRowOff =
        (uint32_t)(n_tokens - 1) * (uint32_t)(EMBED * 4);  // clamp target
#pragma unroll
    for (int k = 0; k < 8; ++k) {
      int i   = tid + k * 256;        // float4 index within tile [0,2048)
      int row = i >> 4;               // 16 float4 per 64-float row
      int col = (i & 15) << 2;        // float column
      // branch-free row clamp (last partial block only); main blocks: no-op
      uint32_t rowOff = (uint32_t)(blockTok + row) * (uint32_t)(EMBED * 4);
      rowOff = min(rowOff, lastRowOff);
      uint64_t gaddr = xbase + rowOff + (uint32_t)(col << 2);
      // low 32 bits of a generic LDS pointer == LDS byte offset (aperture rule)
      uint32_t ldso = (uint32_t)(uintptr_t)&xs[row * ROWPAD + col];
      asm volatile("global_load_async_to_lds_b128 %0, %1, off"
                   :: "v"(ldso), "v"(gaddr) : "memory");
    }
  }

  // ---- preload B fragments (W^T, padded 4->16 cols with zeros), branch-free ----
  // B is 4x16 (KxN) f32 per step: VGPR0 = {K=4s+0 | K=4s+2}, VGPR1 = {K=4s+1 | K=4s+3}
  const int   ncol  = lane & 15;                 // N column this lane owns
  const int   half  = lane >> 4;                 // K sub-pair select
  const int   wrow  = (ncol < NEXP) ? ncol : 0;  // clamped W row
  const float scale = (ncol < NEXP) ? 1.0f : 0.0f;
  const v2f*  wp2   = (const v2f*)(W + wrow * EMBED + 2 * half);
  v2f bfrag[16];
#pragma unroll
  for (int s = 0; s < 16; ++s) {
    v2f t = wp2[2 * s];               // immediate-offset loads off one base
    bfrag[s].x = t.x * scale;
    bfrag[s].y = t.y * scale;
  }

  // wait for this wave's async copies, then block barrier (all waves done)
  asm volatile("s_wait_asynccnt 0" ::: "memory");
  __syncthreads();

  // ---- WMMA chain: 16 steps of 16x16x4 f32 accumulate over K=64 ----
  const int tokBase = wave * TOK_PER_WAVE;       // token tile within block
  const int mrowA   = lane & 15;                 // A-matrix row this lane feeds
  const float* ap = &xs[(tokBase + mrowA) * ROWPAD + 2 * half];

  v8f c = {0.f, 0.f, 0.f, 0.f, 0.f, 0.f, 0.f, 0.f};
#pragma unroll
  for (int s = 0; s < 16; ++s) {
    v2f a = *(const v2f*)(ap + 4 * s);
    // (neg_a, A, neg_b, B, c_mod, C, reuse_a, reuse_b)
    c = __builtin_amdgcn_wmma_f32_16x16x4_f32(
        false, a, false, bfrag[s], (short)0, c, false, false);
  }

  // ---- scatter valid logit columns (experts 0..3) to LDS ----
  if (ncol < NEXP) {
#pragma unroll
    for (int r = 0; r < 8; ++r) {
      int m = half * 8 + r;                       // C layout: M = 8*half + r
      logits_s[(tokBase + m) * NEXP + ncol] = c[r];
    }
  }
  __syncthreads();

  // ---- epilogue: one thread per token: top-2, softmax, dense scatter ----
  if (tid < TOK_PER_BLOCK) {
    int gtok = blockTok + tid;
    if (gtok < n_tokens) {
      float lv0 = logits_s[tid * NEXP + 0];
      float lv1 = logits_s[tid * NEXP + 1];
      float lv2 = logits_s[tid * NEXP + 2];
      float lv3 = logits_s[tid * NEXP + 3];

      // argmax with lowest-index tie-break (jax.lax.top_k semantics)
      int   i0 = 0;  float v0 = lv0;
      if (lv1 > v0) { v0 = lv1; i0 = 1; }
      if (lv2 > v0) { v0 = lv2; i0 = 2; }
      if (lv3 > v0) { v0 = lv3; i0 = 3; }
      int   i1 = -1; float v1 = -INFINITY;
      if (i0 != 0 && lv0 > v1) { v1 = lv0; i1 = 0; }
      if (i0 != 1 && lv1 > v1) { v1 = lv1; i1 = 1; }
      if (i0 != 2 && lv2 > v1) { v1 = lv2; i1 = 2; }
      if (i0 != 3 && lv3 > v1) { v1 = lv3; i1 = 3; }

      // softmax over {v0, v1}; v0 is the max
      float e1  = __expf(v1 - v0);
      float inv = 1.0f / (1.0f + e1);
      float p0  = inv;
      float p1  = e1 * inv;

      // dense scatter without dynamic indexing (avoid scratch)
      float w0 = (i0 == 0) ? p0 : ((i1 == 0) ? p1 : 0.f);
      float w1 = (i0 == 1) ? p0 : ((i1 == 1) ? p1 : 0.f);
      float w2 = (i0 == 2) ? p0 : ((i1 == 2) ? p1 : 0.f);
      float w3 = (i0 == 3) ? p0 : ((i1 == 3) ? p1 : 0.f);

      *(float4*)(weights_out + (size_t)gtok * NEXP) =
          make_float4(w0, w1, w2, w3);
      *(float2*)(idx_out + (size_t)gtok * 2) =
          make_float2((float)i0, (float)i1);
    }
  }
}

extern "C" void kernel_launch(void* const* d_in, const int* in_sizes, int n_in,
                              void* d_out, int out_size, void* d_ws, size_t ws_size,
                              hipStream_t stream) {
  const float* x = (const float*)d_in[0];   // [N, 64]
  const float* W = (const float*)d_in[1];   // [4, 64]
  int n_tokens = in_sizes[0] / EMBED;

  float* weights = (float*)d_out;                              // [N,4]
  float* idx     = (float*)d_out + (size_t)n_tokens * NEXP;    // [N,2]

  int nblocks = (n_tokens + TOK_PER_BLOCK - 1) / TOK_PER_BLOCK;
  moe_gate_wmma_kernel<<<nblocks, 256, 0, stream>>>(x, W, weights, idx, n_tokens);
}